// RGATLayer_45114336477932
// MI455X (gfx1250) — compile-verified
//
#include <hip/hip_runtime.h>

#define NN 50000          // nodes
#define NE 800000         // edges (without self loops)
#define KDIM 256          // in_dim
#define ODIM 256          // heads*head_dim
#define NH 4              // heads
#define HC 64             // head_dim
#define E2 (NE + NN)      // edges + self loops
#define NEG_SLOPE 0.2f
#define EPS_F 1e-16f

typedef float v2f __attribute__((ext_vector_type(2)));
typedef float v8f __attribute__((ext_vector_type(8)));

// ---------- order-preserving float <-> uint map for atomic max ----------
__device__ __forceinline__ unsigned f2mono(float f) {
    int i = __float_as_int(f);
    return (i >= 0) ? ((unsigned)i | 0x80000000u) : ~(unsigned)i;
}
__device__ __forceinline__ float mono2f(unsigned u) {
    unsigned i = (u & 0x80000000u) ? (u & 0x7fffffffu) : ~u;
    return __uint_as_float(i);
}
__device__ __forceinline__ float lrelu(float v) {
    return v > 0.0f ? v : NEG_SLOPE * v;
}

__device__ __forceinline__ void edge_sd(const long long* __restrict__ ei, int e,
                                        int& s, int& d) {
    if (e < NE) { s = (int)ei[e]; d = (int)ei[NE + e]; }
    else        { s = d = e - NE; }               // self loop
}

// ---------- K1: H = X * W via V_WMMA_F32_16X16X4_F32, B staged in LDS ----------
// block = 256 threads = 8 waves; wave w -> rows [bx*128 + 16w, +16), cols [by*64, +64)
// W strip (256x64, 64KB) is staged once per block into LDS pre-swizzled so each
// B fragment is one conflict-free ds_load_b64 per lane.
__global__ __launch_bounds__(256) void gat_gemm(const float* __restrict__ X,
                                                const float* __restrict__ W,
                                                float* __restrict__ Hh) {
    __shared__ float ldsB[64 * 4 * 32 * 2];   // [kk][j][lane][2] : 64 KB

    const int tid   = threadIdx.x;
    const int lane  = tid & 31;
    const int wave  = tid >> 5;
    const int laneM = lane & 15;
    const int half  = lane >> 4;          // 0: lanes 0-15, 1: lanes 16-31
    const int rowBase = blockIdx.x * 128 + wave * 16;
    const int colBase = blockIdx.y * 64;

    // ---- cooperative stage of W[0:256][colBase:+64] in fragment order ----
    {
        const int c  = tid & 63;          // column in strip (coalesced global read)
        const int j  = c >> 4;            // n-subtile 0..3
        const int m  = c & 15;
        for (int r = (tid >> 6); r < KDIM; r += 4) {
            const int kk = r >> 2;        // k-step
            const int h  = (r >> 1) & 1;  // lane half that consumes this row
            const int p  = r & 1;         // slot within the b64 fragment
            ldsB[(((kk * 4 + j) * 32) + (h * 16 + m)) * 2 + p] =
                W[(size_t)r * ODIM + colBase + c];
        }
    }
    __syncthreads();

    // clamp load row (stores are guarded); WMMA requires EXEC all-ones
    const int rowL = min(rowBase + laneM, NN - 1);
    const float* __restrict__ xrow = X + (size_t)rowL * KDIM;

    v8f acc0 = {}, acc1 = {}, acc2 = {}, acc3 = {};

    for (int kk = 0; kk < KDIM / 4; ++kk) {
        // A 16x4: VGPR0 = K {0|2}, VGPR1 = K {1|3} by lane half -> one b64 load
        const float2 av = *(const float2*)(xrow + 4 * kk + 2 * half);
        v2f a; a.x = av.x; a.y = av.y;
        // B fragments: single ds_load_b64 each, conflict-free
        const float* bb = &ldsB[(kk * 4 * 32 + lane) * 2];
        v2f b0 = *(const v2f*)(bb + 0 * 64);
        v2f b1 = *(const v2f*)(bb + 1 * 64);
        v2f b2 = *(const v2f*)(bb + 2 * 64);
        v2f b3 = *(const v2f*)(bb + 3 * 64);
        acc0 = __builtin_amdgcn_wmma_f32_16x16x4_f32(false, a, false, b0, (short)0, acc0, false, false);
        acc1 = __builtin_amdgcn_wmma_f32_16x16x4_f32(false, a, false, b1, (short)0, acc1, false, false);
        acc2 = __builtin_amdgcn_wmma_f32_16x16x4_f32(false, a, false, b2, (short)0, acc2, false, false);
        acc3 = __builtin_amdgcn_wmma_f32_16x16x4_f32(false, a, false, b3, (short)0, acc3, false, false);
    }

    // C/D layout: VGPR r -> M = r + 8*half, N = laneM
    #pragma unroll
    for (int r = 0; r < 8; ++r) {
        int row = rowBase + r + 8 * half;
        if (row < NN) {
            float* __restrict__ orow = Hh + (size_t)row * ODIM + colBase + laneM;
            orow[0]  = acc0[r];
            orow[16] = acc1[r];
            orow[32] = acc2[r];
            orow[48] = acc3[r];
        }
    }
}

// ---------- K2: per-(node,head) attention dot products ----------
__global__ __launch_bounds__(256) void gat_alpha(const float* __restrict__ Hh,
                                                 const float* __restrict__ att_src,
                                                 const float* __restrict__ att_dst,
                                                 float* __restrict__ as_,
                                                 float* __restrict__ ad_) {
    const int lane = threadIdx.x & 31;
    const int wave = threadIdx.x >> 5;
    const int flat = blockIdx.x * 8 + wave;        // (node, head) id — uniform per wave
    if (flat >= NN * NH) return;
    const int n = flat >> 2, hd = flat & 3;
    const float* __restrict__ hp  = Hh + (size_t)n * ODIM + hd * HC;
    const float* __restrict__ asv = att_src + hd * HC;
    const float* __restrict__ adv = att_dst + hd * HC;
    float ps = hp[lane] * asv[lane] + hp[lane + 32] * asv[lane + 32];
    float pd = hp[lane] * adv[lane] + hp[lane + 32] * adv[lane + 32];
    #pragma unroll
    for (int off = 16; off > 0; off >>= 1) {
        ps += __shfl_xor(ps, off, 32);
        pd += __shfl_xor(pd, off, 32);
    }
    if (lane == 0) { as_[flat] = ps; ad_[flat] = pd; }
}

// ---------- K3: init out=bias, max buffer=sentinel(0), sum=0 ----------
__global__ void gat_init(float* __restrict__ out, const float* __restrict__ bias,
                         unsigned* __restrict__ mU, float* __restrict__ sS) {
    int idx = blockIdx.x * blockDim.x + threadIdx.x;
    if (idx < NN * ODIM) out[idx] = bias[idx & (ODIM - 1)];
    if (idx < NN * NH)   { mU[idx] = 0u; sS[idx] = 0.0f; }
}

// ---------- K4: segment max of leaky-relu logits (one thread = one edge, 4 heads) ----------
__global__ void gat_emax(const long long* __restrict__ ei,
                         const float* __restrict__ as_, const float* __restrict__ ad_,
                         unsigned* __restrict__ mU) {
    int e = blockIdx.x * blockDim.x + threadIdx.x;
    if (e >= E2) return;
    int s, d; edge_sd(ei, e, s, d);
    const float4 vs = *(const float4*)(as_ + (size_t)s * NH);
    const float4 vd = *(const float4*)(ad_ + (size_t)d * NH);
    unsigned* __restrict__ mp = mU + (size_t)d * NH;
    atomicMax(mp + 0, f2mono(lrelu(vs.x + vd.x)));
    atomicMax(mp + 1, f2mono(lrelu(vs.y + vd.y)));
    atomicMax(mp + 2, f2mono(lrelu(vs.z + vd.z)));
    atomicMax(mp + 3, f2mono(lrelu(vs.w + vd.w)));
}

// ---------- K5: exp(e - max), segment sum (one thread = one edge, 4 heads) ----------
__global__ void gat_esum(const long long* __restrict__ ei,
                         const float* __restrict__ as_, const float* __restrict__ ad_,
                         const unsigned* __restrict__ mU,
                         float* __restrict__ sS, float* __restrict__ exb) {
    int e = blockIdx.x * blockDim.x + threadIdx.x;
    if (e >= E2) return;
    int s, d; edge_sd(ei, e, s, d);
    const float4 vs = *(const float4*)(as_ + (size_t)s * NH);
    const float4 vd = *(const float4*)(ad_ + (size_t)d * NH);
    const uint4  mu = *(const uint4*)(mU + (size_t)d * NH);
    float4 ex;
    ex.x = __expf(lrelu(vs.x + vd.x) - mono2f(mu.x));
    ex.y = __expf(lrelu(vs.y + vd.y) - mono2f(mu.y));
    ex.z = __expf(lrelu(vs.z + vd.z) - mono2f(mu.z));
    ex.w = __expf(lrelu(vs.w + vd.w) - mono2f(mu.w));
    *(float4*)(exb + (size_t)e * NH) = ex;
    float* __restrict__ sp = sS + (size_t)d * NH;
    atomicAdd(sp + 0, ex.x);
    atomicAdd(sp + 1, ex.y);
    atomicAdd(sp + 2, ex.z);
    atomicAdd(sp + 3, ex.w);
}

// ---------- K6: weighted scatter-add of messages (64 lanes x float4 per edge) ----------
__global__ __launch_bounds__(256) void gat_aggr(const long long* __restrict__ ei,
                                                const float* __restrict__ Hh,
                                                const float* __restrict__ exb,
                                                const float* __restrict__ sS,
                                                float* __restrict__ out) {
    __shared__ int sd[4][2];                   // (src,dst) for the 4 edges of this block
    const int eslot = threadIdx.x >> 6;
    const int e     = blockIdx.x * 4 + eslot;
    if (threadIdx.x < 4) {
        int ee = blockIdx.x * 4 + threadIdx.x;
        int s = 0, d = 0;
        if (ee < E2) edge_sd(ei, ee, s, d);
        sd[threadIdx.x][0] = s;
        sd[threadIdx.x][1] = d;
    }
    __syncthreads();
    if (e >= E2) return;
    const int s  = sd[eslot][0];
    const int d  = sd[eslot][1];
    const int c  = threadIdx.x & 63;           // 4-channel group
    const int cb = c << 2;                     // channel base 0..252, 16B aligned
    const int hd = cb >> 6;                    // head index
    const float alpha = exb[(size_t)e * NH + hd] / (sS[(size_t)d * NH + hd] + EPS_F);
    const float4 hv = *(const float4*)(Hh + (size_t)s * ODIM + cb);
    float* __restrict__ op = out + (size_t)d * ODIM + cb;
    atomicAdd(op + 0, hv.x * alpha);
    atomicAdd(op + 1, hv.y * alpha);
    atomicAdd(op + 2, hv.z * alpha);
    atomicAdd(op + 3, hv.w * alpha);
}

extern "C" void kernel_launch(void* const* d_in, const int* in_sizes, int n_in,
                              void* d_out, int out_size, void* d_ws, size_t ws_size,
                              hipStream_t stream) {
    const float*     x       = (const float*)d_in[0];
    const long long* ei      = (const long long*)d_in[1];   // int64 edge_index [2,E]
    const float*     W       = (const float*)d_in[2];
    const float*     att_src = (const float*)d_in[3];
    const float*     att_dst = (const float*)d_in[4];
    const float*     bias    = (const float*)d_in[5];
    float*           out     = (float*)d_out;

    char* ws = (char*)d_ws;
    float*    Hh  = (float*)ws;    ws += (size_t)NN * ODIM * sizeof(float);  // 51.2 MB
    float*    as_ = (float*)ws;    ws += (size_t)NN * NH * sizeof(float);
    float*    ad_ = (float*)ws;    ws += (size_t)NN * NH * sizeof(float);
    unsigned* mU  = (unsigned*)ws; ws += (size_t)NN * NH * sizeof(unsigned);
    float*    sS  = (float*)ws;    ws += (size_t)NN * NH * sizeof(float);
    float*    exb = (float*)ws;    // E2 * NH floats (13.6 MB)

    dim3 gGemm((NN + 127) / 128, ODIM / 64);   // 391 x 4
    gat_gemm<<<gGemm, 256, 0, stream>>>(x, W, Hh);
    gat_alpha<<<(NN * NH + 7) / 8, 256, 0, stream>>>(Hh, att_src, att_dst, as_, ad_);
    gat_init<<<(NN * ODIM + 255) / 256, 256, 0, stream>>>(out, bias, mU, sS);
    gat_emax<<<(E2 + 255) / 256, 256, 0, stream>>>(ei, as_, ad_, mU);
    gat_esum<<<(E2 + 255) / 256, 256, 0, stream>>>(ei, as_, ad_, mU, sS, exb);
    gat_aggr<<<(E2 + 3) / 4, 256, 0, stream>>>(ei, Hh, exb, sS, out);
}